// GCNNetwork_3435973837102
// MI455X (gfx1250) — compile-verified
//
#include <hip/hip_runtime.h>
#include <hip/hip_bf16.h>

#define D_OUT 64
#define D_INN 92
#define BN_EPS 1e-5f

typedef __attribute__((ext_vector_type(2))) float v2f;
typedef __attribute__((ext_vector_type(8))) float v8f;
typedef __attribute__((__vector_size__(16))) int vi4;

#ifndef __has_builtin
#define __has_builtin(x) 0
#endif
#if __has_builtin(__builtin_amdgcn_global_load_async_to_lds_b128) && \
    __has_builtin(__builtin_amdgcn_s_wait_asynccnt)
#define USE_ASYNC_LDS 1
#else
#define USE_ASYNC_LDS 0
#endif

// 16-byte global -> LDS copy; async (ASYNCcnt-tracked, bypasses VGPRs) when available
__device__ __forceinline__ void copy16_to_lds(const float* __restrict__ g, float* l) {
#if USE_ASYNC_LDS
  __builtin_amdgcn_global_load_async_to_lds_b128(
      (__attribute__((address_space(1))) vi4*)g,
      (__attribute__((address_space(3))) vi4*)l, 0, 0);
#else
  *(float4*)l = *(const float4*)g;
#endif
}
__device__ __forceinline__ void lds_copy_fence() {
#if USE_ASYNC_LDS
  __builtin_amdgcn_s_wait_asynccnt(0);
#endif
}

// ---------------- utility kernels ----------------
__global__ void zero_f(float* __restrict__ p, long long n) {
  long long i = blockIdx.x * (long long)blockDim.x + threadIdx.x;
  if (i < n) p[i] = 0.0f;
}

__global__ void deg_init(float* __restrict__ deg, int n) {
  int i = blockIdx.x * blockDim.x + threadIdx.x;
  if (i < n) deg[i] = 1.0f;  // self-loop weight
}

__global__ void deg_acc(const int* __restrict__ col, const float* __restrict__ w,
                        float* __restrict__ deg, int e) {
  int i = blockIdx.x * blockDim.x + threadIdx.x;
  if (i < e) atomicAdd(&deg[col[i]], w[i]);
}

__global__ void dinv_k(float* __restrict__ deg, int n) {
  int i = blockIdx.x * blockDim.x + threadIdx.x;
  if (i < n) {
    float d = deg[i];
    deg[i] = d > 0.0f ? rsqrtf(d) : 0.0f;
  }
}

__global__ void norm_k(const int* __restrict__ row, const int* __restrict__ col,
                       const float* __restrict__ w, const float* __restrict__ dinv,
                       float* __restrict__ norm, int e) {
  int i = blockIdx.x * blockDim.x + threadIdx.x;
  if (i < e) norm[i] = dinv[row[i]] * w[i] * dinv[col[i]];
}

// ---------------- fp32 WMMA GEMM: H[N,64] = X[N,K] @ W[K,64] ----------------
// Persistent: 128 threads = 4 waves per block; W panel staged in LDS once per
// block, then grid-stride over 16-row tiles. K in {92,64}, both multiples of 4
// -> exact 16x16x4 fp32 WMMA tiling. A/W staged via async global->LDS copies.
__global__ __launch_bounds__(128) void gemm_wmma_f32(
    const float* __restrict__ X, const float* __restrict__ W,
    float* __restrict__ H, int K, int numTiles) {
  __shared__ float sB[D_INN * D_OUT];  // K x 64 weights (row-major)
  __shared__ float sA[16 * D_INN];     // 16 x K tile of X (row-major, stride K)
  const int tid = threadIdx.x;

  // stage weight panel once per block (16B chunks; K*64 is a multiple of 4)
  for (int idx = tid * 4; idx < K * D_OUT; idx += 128 * 4)
    copy16_to_lds(W + idx, sB + idx);

  const int lane = tid & 31;
  const int wave = tid >> 5;
  const int n0 = wave * 16;          // output column strip for this wave
  const int lm = lane & 15;          // row (A) / col (B,C) within the tile
  const int kh = (lane >> 4) << 1;   // lanes 0-15 -> K {0,1}; 16-31 -> {2,3}

  for (int t = blockIdx.x; t < numTiles; t += gridDim.x) {
    const long long m0 = (long long)t * 16;
    // rows m0..m0+15 are contiguous in X (16*K floats, 16B-aligned chunks)
    for (int idx = tid * 4; idx < 16 * K; idx += 128 * 4)
      copy16_to_lds(X + m0 * K + idx, sA + idx);
    lds_copy_fence();
    __syncthreads();

    v8f acc = {};
    for (int k = 0; k < K; k += 4) {
      v2f a, b;
      a.x = sA[lm * K + k + kh];
      a.y = sA[lm * K + k + kh + 1];
      b.x = sB[(k + kh) * D_OUT + n0 + lm];
      b.y = sB[(k + kh + 1) * D_OUT + n0 + lm];
      acc = __builtin_amdgcn_wmma_f32_16x16x4_f32(
          false, a, false, b, (short)0, acc, false, false);
    }

    // D layout: VGPR r -> row m0+r (lanes 0-15) / m0+r+8 (lanes 16-31)
    const long long rbase = m0 + ((lane >> 4) << 3);
    const int nc = n0 + lm;
#pragma unroll
    for (int r = 0; r < 8; ++r)
      H[(rbase + r) * D_OUT + nc] = acc[r];
    __syncthreads();  // all waves done with sA before next tile overwrites it
  }
}

// ---------------- message passing ----------------
// OUT[i,:] = dinv[i]^2 * H[i,:]   (self-loop term, also initializes OUT)
__global__ void selfloop_init(const float* __restrict__ H,
                              const float* __restrict__ dinv,
                              float* __restrict__ OUT, long long n64) {
  long long idx = blockIdx.x * (long long)blockDim.x + threadIdx.x;
  if (idx < n64) {
    float dv = dinv[idx >> 6];
    OUT[idx] = dv * dv * H[idx];
  }
}

// OUT[col,:] += norm[e] * H[row,:]  (16 threads/edge, float4 per thread)
__global__ void edge_scatter(const int* __restrict__ row, const int* __restrict__ col,
                             const float* __restrict__ norm, const float* __restrict__ H,
                             float* __restrict__ OUT, long long total) {
  long long idx = blockIdx.x * (long long)blockDim.x + threadIdx.x;
  if (idx >= total) return;
  long long e = idx >> 4;
  int c4 = (int)(idx & 15) << 2;
  float nv = norm[e];
  float4 h4 = *(const float4*)(H + (long long)row[e] * D_OUT + c4);
  float* op = OUT + (long long)col[e] * D_OUT + c4;
  atomicAdd(op + 0, nv * h4.x);
  atomicAdd(op + 1, nv * h4.y);
  atomicAdd(op + 2, nv * h4.z);
  atomicAdd(op + 3, nv * h4.w);
}

// ---------------- batchnorm (training mode) ----------------
__global__ __launch_bounds__(256) void bn_stats(const float* __restrict__ H,
                                                float* __restrict__ stats, int n) {
  const int c = threadIdx.x & 63;
  const int sub = threadIdx.x >> 6;  // 0..3
  float s = 0.0f, ss = 0.0f;
  for (int i = blockIdx.x * 4 + sub; i < n; i += gridDim.x * 4) {
    float v = H[(long long)i * D_OUT + c];
    s += v;
    ss += v * v;
  }
  __shared__ float red[512];
  red[threadIdx.x] = s;
  red[256 + threadIdx.x] = ss;
  __syncthreads();
  if (sub == 0) {
    for (int j = 1; j < 4; ++j) {
      s += red[j * 64 + c];
      ss += red[256 + j * 64 + c];
    }
    atomicAdd(&stats[c], s);
    atomicAdd(&stats[64 + c], ss);
  }
}

__global__ void bn_apply_relu(float* __restrict__ H, const float* __restrict__ stats,
                              const float* __restrict__ g, const float* __restrict__ beta,
                              int n, long long n64) {
  long long idx = blockIdx.x * (long long)blockDim.x + threadIdx.x;
  if (idx >= n64) return;
  int c = (int)(idx & 63);
  float invN = 1.0f / (float)n;
  float mu = stats[c] * invN;
  float var = stats[64 + c] * invN - mu * mu;  // biased, matches jnp.var
  float v = (H[idx] - mu) * rsqrtf(var + BN_EPS) * g[c] + beta[c];
  H[idx] = v > 0.0f ? v : 0.0f;
}

// ---------------- pooling + MLP ----------------
__global__ void pool_scatter(const float* __restrict__ H, const int* __restrict__ batch,
                             float* __restrict__ pooled, float* __restrict__ cnt,
                             long long n64) {
  long long idx = blockIdx.x * (long long)blockDim.x + threadIdx.x;
  if (idx >= n64) return;
  int i = (int)(idx >> 6);
  int c = (int)(idx & 63);
  int gph = batch[i];
  atomicAdd(&pooled[(long long)gph * D_OUT + c], H[idx]);
  if (c == 0) atomicAdd(&cnt[gph], 1.0f);
}

__global__ void mlp1(const float* __restrict__ pooled, const float* __restrict__ cnt,
                     const float* __restrict__ Wp1, const float* __restrict__ bp1,
                     float* __restrict__ t, int G) {
  int idx = blockIdx.x * blockDim.x + threadIdx.x;
  if (idx >= G * 100) return;
  int gph = idx / 100, j = idx % 100;
  float inv = 1.0f / fmaxf(cnt[gph], 1.0f);
  float acc = bp1[j];
  for (int k = 0; k < D_OUT; ++k)
    acc += pooled[gph * D_OUT + k] * inv * Wp1[k * 100 + j];
  t[idx] = fmaxf(acc, 0.0f);
}

__global__ void mlp2(const float* __restrict__ t, const float* __restrict__ Wp2,
                     const float* __restrict__ bp2, float* __restrict__ out, int G) {
  int gph = blockIdx.x * blockDim.x + threadIdx.x;
  if (gph >= G) return;
  float acc = bp2[0];
  for (int j = 0; j < 100; ++j) acc += t[gph * 100 + j] * Wp2[j];
  out[gph] = acc;
}

// ---------------- host launcher ----------------
extern "C" void kernel_launch(void* const* d_in, const int* in_sizes, int n_in,
                              void* d_out, int out_size, void* d_ws, size_t ws_size,
                              hipStream_t stream) {
  const float* x     = (const float*)d_in[0];
  const int*   ei    = (const int*)d_in[1];
  const float* ew    = (const float*)d_in[2];
  const int*   batch = (const int*)d_in[3];
  const float* W0 = (const float*)d_in[4];
  const float* g0 = (const float*)d_in[6];
  const float* beta0 = (const float*)d_in[7];
  const float* W1 = (const float*)d_in[8];
  const float* g1 = (const float*)d_in[10];
  const float* beta1 = (const float*)d_in[11];
  const float* W2 = (const float*)d_in[12];
  const float* g2 = (const float*)d_in[14];
  const float* beta2 = (const float*)d_in[15];
  const float* Wp1 = (const float*)d_in[16];
  const float* bp1 = (const float*)d_in[17];
  const float* Wp2 = (const float*)d_in[18];
  const float* bp2 = (const float*)d_in[19];
  float* out = (float*)d_out;

  const int N = in_sizes[0] / D_INN;  // 100000
  const int E = in_sizes[2];          // 1600000
  const int G = out_size;             // 256
  const int* row = ei;
  const int* col = ei + E;

  // workspace carve-up (floats)
  float* ws = (float*)d_ws;
  float* buf0   = ws;                          // N*64  (GEMM output)
  float* buf1   = buf0 + (size_t)N * D_OUT;    // N*64  (scatter/activation)
  float* dinv   = buf1 + (size_t)N * D_OUT;    // N     (deg -> deg^-1/2 in place)
  float* norm   = dinv + N;                    // E
  float* stats  = norm + E;                    // 128
  float* pooled = stats + 128;                 // G*64
  float* cnt    = pooled + (size_t)G * D_OUT;  // G
  float* tbuf   = cnt + G;                     // G*100

  const long long n64 = (long long)N * D_OUT;
  const int TB = 256;
  const int gN   = (N + TB - 1) / TB;
  const int gE   = (E + TB - 1) / TB;
  const int gN64 = (int)((n64 + TB - 1) / TB);
  const long long escat = (long long)E * 16;
  const int gES = (int)((escat + TB - 1) / TB);
  const int numTiles = N / 16;
  const int gemmBlocks = numTiles < 1024 ? numTiles : 1024;

  // --- gcn_norm: degrees (with self-loops), dinv, per-edge norm ---
  deg_init<<<gN, TB, 0, stream>>>(dinv, N);
  deg_acc<<<gE, TB, 0, stream>>>(col, ew, dinv, E);
  dinv_k<<<gN, TB, 0, stream>>>(dinv, N);
  norm_k<<<gE, TB, 0, stream>>>(row, col, ew, dinv, norm, E);

  const float* in_feat = x;
  int K = D_INN;
  const float* Wl[3] = {W0, W1, W2};
  const float* Gl[3] = {g0, g1, g2};
  const float* Bl[3] = {beta0, beta1, beta2};

  for (int l = 0; l < 3; ++l) {
    gemm_wmma_f32<<<gemmBlocks, 128, 0, stream>>>(in_feat, Wl[l], buf0, K, numTiles);
    selfloop_init<<<gN64, TB, 0, stream>>>(buf0, dinv, buf1, n64);
    edge_scatter<<<gES, TB, 0, stream>>>(row, col, norm, buf0, buf1, escat);
    zero_f<<<1, 128, 0, stream>>>(stats, 128);
    bn_stats<<<256, 256, 0, stream>>>(buf1, stats, N);
    bn_apply_relu<<<gN64, TB, 0, stream>>>(buf1, stats, Gl[l], Bl[l], N, n64);
    in_feat = buf1;
    K = D_OUT;
  }

  // --- global mean pool + MLP head ---
  zero_f<<<(G * D_OUT + G + 127) / 128, 128, 0, stream>>>(pooled, (long long)G * D_OUT + G);
  pool_scatter<<<gN64, TB, 0, stream>>>(buf1, batch, pooled, cnt, n64);
  mlp1<<<(G * 100 + TB - 1) / TB, TB, 0, stream>>>(pooled, cnt, Wp1, bp1, tbuf, G);
  mlp2<<<(G + TB - 1) / TB, TB, 0, stream>>>(tbuf, Wp2, bp2, out, G);
}